// DGMC_55130200211623
// MI455X (gfx1250) — compile-verified
//
#include <hip/hip_runtime.h>

// ---------------- problem constants ----------------
#define Bsz   4
#define Nn    1024
#define Ff    128
#define Rr    16
#define Ee    65536          // B*N*DEG
#define BN    4096           // B*N
#define NSTEPS 2
#define SHATSZ (4u*1024u*1024u)

typedef __attribute__((ext_vector_type(2))) float v2f;
typedef __attribute__((ext_vector_type(8))) float v8f;

__device__ __forceinline__ v8f wmma4(v2f a, v2f b, v8f c) {
  // D = A(16x4 f32) * B(4x16 f32) + C(16x16 f32)
  return __builtin_amdgcn_wmma_f32_16x16x4_f32(false, a, false, b, (short)0, c,
                                               false, false);
}

// async global -> LDS copy, 16 bytes per lane (ASYNCcnt-tracked)
__device__ __forceinline__ void async_cp16(unsigned lds_off, const float* gptr) {
  asm volatile("global_load_async_to_lds_b128 %0, %1, off"
               :: "v"(lds_off), "v"(gptr) : "memory");
}
__device__ __forceinline__ void async_wait0() {
  asm volatile("s_wait_asynccnt 0" ::: "memory");
}

// ---------------- utility kernels ----------------
__global__ void k_zero(float* __restrict__ p, int n) {
  int i = blockIdx.x * 256 + threadIdx.x;
  if (i < n) p[i] = 0.f;
}

// scatter-add x[src] into agg[dst], F=128 features  (one thread per (edge,feat))
__global__ void k_scatter128(const float* __restrict__ x, const int* __restrict__ ei,
                             float* __restrict__ agg) {
  unsigned tid = blockIdx.x * 256 + threadIdx.x;   // Ee*128 threads
  unsigned e = tid >> 7, f = tid & 127;
  int src = ei[e];
  int dst = ei[Ee + e];
  atomicAdd(&agg[(size_t)dst * 128 + f], x[(size_t)src * 128 + f]);
}

// scatter-add, F=16
__global__ void k_scatter16(const float* __restrict__ x, const int* __restrict__ ei,
                            float* __restrict__ agg) {
  unsigned tid = blockIdx.x * 256 + threadIdx.x;   // Ee*16 threads
  unsigned e = tid >> 4, f = tid & 15;
  int src = ei[e];
  int dst = ei[Ee + e];
  atomicAdd(&agg[(size_t)dst * 16 + f], x[(size_t)src * 16 + f]);
}

// ---------------- psi_1 : H = relu(X@Ws + AGG@Wn + b)   M=4096,N=128,K=128 ----
__global__ void k_psi1_gemm(const float* __restrict__ X, const float* __restrict__ AGG,
                            const float* __restrict__ Ws, const float* __restrict__ Wn,
                            const float* __restrict__ bias, float* __restrict__ H) {
  int wave = threadIdx.x >> 5, lane = threadIdx.x & 31;
  int tile = blockIdx.x * 8 + wave;        // 2048 tiles = 256(M) x 8(N)
  int tm = (tile >> 3) << 4;
  int tn = (tile & 7) << 4;
  int m = tm + (lane & 15);
  int n = tn + (lane & 15);
  int kh = (lane >> 4) << 1;
  v8f acc = {};
  for (int k = 0; k < 128; k += 4) {
    int kb = k + kh;
    v2f a, b;
    a.x = X[(size_t)m * 128 + kb];     a.y = X[(size_t)m * 128 + kb + 1];
    b.x = Ws[(size_t)kb * 128 + n];    b.y = Ws[(size_t)(kb + 1) * 128 + n];
    acc = wmma4(a, b, acc);
    a.x = AGG[(size_t)m * 128 + kb];   a.y = AGG[(size_t)m * 128 + kb + 1];
    b.x = Wn[(size_t)kb * 128 + n];    b.y = Wn[(size_t)(kb + 1) * 128 + n];
    acc = wmma4(a, b, acc);
  }
  float bn = bias[n];
  int mb = tm + ((lane >> 4) << 3);
#pragma unroll
  for (int i = 0; i < 8; ++i) {
    float v = acc[i] + bn;
    H[(size_t)(mb + i) * 128 + n] = v > 0.f ? v : 0.f;
  }
}

// ---------------- S_hat[b] = h_s[b] @ h_t[b]^T    per batch 1024x1024, K=128 --
// LDS-tiled: block = 256 threads computes a 64x64 output tile; A/B K-slabs are
// staged via async global->LDS (2 phases of K=64). LDA=68 pads rows so the
// 16-lane x stride access pattern is LDS-bank conflict-free (stride%64 == 4).
#define LDA 68
__global__ void k_shat_gemm(const float* __restrict__ hs, const float* __restrict__ ht,
                            float* __restrict__ Shat) {
  __shared__ float As[64 * LDA];
  __shared__ float Bs[64 * LDA];
  int tid = threadIdx.x;
  int wave = tid >> 5, lane = tid & 31;
  int blk = blockIdx.x;                  // 1024 blocks = 4(b) x 16 x 16
  int b  = blk >> 8;
  int m0 = ((blk >> 4) & 15) << 6;       // 64-row slab of s
  int n0 = (blk & 15) << 6;              // 64-row slab of t
  const float* Ag = hs + (size_t)b * Nn * Ff + (size_t)m0 * Ff;
  const float* Bg = ht + (size_t)b * Nn * Ff + (size_t)n0 * Ff;
  unsigned aBase = (unsigned)(uintptr_t)As;   // AS3 offset lives in low 32 bits
  unsigned bBase = (unsigned)(uintptr_t)Bs;

  // each wave: two 16x16 tiles (mt, nt) and (mt, nt+16), sharing the A fragment
  int mt = (wave & 3) << 4;              // 0,16,32,48
  int nt = (wave >> 2) << 5;             // 0,32
  int mi = mt + (lane & 15);
  int ni = nt + (lane & 15);
  int kh = (lane >> 4) << 1;
  v8f acc0 = {}, acc1 = {};

  for (int kk = 0; kk < 128; kk += 64) {
    // stage 64 rows x 64 k-cols of A and B: 1024 float4 chunks each, 4/thread
#pragma unroll
    for (int j = 0; j < 4; ++j) {
      int c = tid + (j << 8);
      int row = c >> 4, col = (c & 15) << 2;
      unsigned loff = (unsigned)(row * LDA + col) * 4u;
      async_cp16(aBase + loff, Ag + (size_t)row * Ff + kk + col);
      async_cp16(bBase + loff, Bg + (size_t)row * Ff + kk + col);
    }
    async_wait0();
    __syncthreads();
#pragma unroll
    for (int k = 0; k < 64; k += 4) {
      int kb = k + kh;
      v2f a  = *(const v2f*)&As[mi * LDA + kb];
      v2f b0 = *(const v2f*)&Bs[ni * LDA + kb];
      v2f b1 = *(const v2f*)&Bs[(ni + 16) * LDA + kb];
      acc0 = wmma4(a, b0, acc0);
      acc1 = wmma4(a, b1, acc1);
    }
    __syncthreads();   // protect LDS before next stage phase
  }

  float* outp = Shat + (size_t)b * Nn * Nn;
  int mb = m0 + mt + ((lane >> 4) << 3);
  int nn = n0 + nt + (lane & 15);
#pragma unroll
  for (int i = 0; i < 8; ++i) {
    outp[(size_t)(mb + i) * Nn + nn]      = acc0[i];
    outp[(size_t)(mb + i) * Nn + nn + 16] = acc1[i];
  }
}

// ---------------- row softmax stats: max and 1/sum(exp) ----------------------
__global__ void k_row_stats(const float* __restrict__ Shat, float* __restrict__ mx,
                            float* __restrict__ sinv) {
  __shared__ float red[256];
  int row = blockIdx.x;                    // 4096 rows
  const float* p = Shat + (size_t)row * Nn;
  float m = -1e30f;
  for (int i = threadIdx.x; i < Nn; i += 256) m = fmaxf(m, p[i]);
  red[threadIdx.x] = m;
  __syncthreads();
  for (int s = 128; s > 0; s >>= 1) {
    if (threadIdx.x < s) red[threadIdx.x] = fmaxf(red[threadIdx.x], red[threadIdx.x + s]);
    __syncthreads();
  }
  float rm = red[0];
  __syncthreads();
  float sum = 0.f;
  for (int i = threadIdx.x; i < Nn; i += 256) sum += __expf(p[i] - rm);
  red[threadIdx.x] = sum;
  __syncthreads();
  for (int s = 128; s > 0; s >>= 1) {
    if (threadIdx.x < s) red[threadIdx.x] += red[threadIdx.x + s];
    __syncthreads();
  }
  if (threadIdx.x == 0) { mx[row] = rm; sinv[row] = 1.0f / red[0]; }
}

// ---------------- softmax write-out ------------------------------------------
__global__ void k_softmax_out(const float* __restrict__ Shat, const float* __restrict__ mx,
                              const float* __restrict__ sinv, float* __restrict__ out) {
  size_t idx = (size_t)blockIdx.x * 256 + threadIdx.x;   // 4M elements
  int row = (int)(idx >> 10);
  out[idx] = __expf(Shat[idx] - mx[row]) * sinv[row];
}

// ---------------- r_t[b] = softmax(S_hat[b])^T @ r_s[b]   M=1024,N=16,K=1024 --
__global__ void k_rt_gemm(const float* __restrict__ Shat, const float* __restrict__ mx,
                          const float* __restrict__ sinv, const float* __restrict__ rs,
                          float* __restrict__ rt) {
  int wave = threadIdx.x >> 5, lane = threadIdx.x & 31;
  int tile = blockIdx.x * 8 + wave;     // 256 tiles = 4(b) x 64(t-tiles)
  int b = tile >> 6;
  int tm = (tile & 63) << 4;            // t dimension
  const float* S   = Shat + (size_t)b * Nn * Nn;
  const float* mxb = mx + b * Nn;
  const float* svb = sinv + b * Nn;
  const float* rsb = rs + (size_t)b * Nn * Rr;
  int m = tm + (lane & 15);             // t index
  int n = lane & 15;                    // r index
  int kh = (lane >> 4) << 1;
  v8f acc = {};
  for (int k = 0; k < Nn; k += 4) {
    int kb = k + kh;                    // s index
    v2f a, bb;
    a.x = __expf(S[(size_t)kb * Nn + m] - mxb[kb]) * svb[kb];
    a.y = __expf(S[(size_t)(kb + 1) * Nn + m] - mxb[kb + 1]) * svb[kb + 1];
    bb.x = rsb[(size_t)kb * Rr + n];
    bb.y = rsb[(size_t)(kb + 1) * Rr + n];
    acc = wmma4(a, bb, acc);
  }
  float* out = rt + (size_t)b * Nn * Rr;
  int mb = tm + ((lane >> 4) << 3);
#pragma unroll
  for (int i = 0; i < 8; ++i) out[(size_t)(mb + i) * Rr + (lane & 15)] = acc[i];
}

// ---------------- generic 16-wide GEMM: C = f(A@Wa [+ A2@Wb] + bias) ---------
// M=4096, N=16, K=16
__global__ void k_gemm16(const float* __restrict__ A, const float* __restrict__ Wa,
                         const float* __restrict__ A2, const float* __restrict__ Wb,
                         const float* __restrict__ bias, float* __restrict__ C,
                         int do_relu) {
  int wave = threadIdx.x >> 5, lane = threadIdx.x & 31;
  int tm = (blockIdx.x * 8 + wave) << 4;     // 256 tiles
  int m = tm + (lane & 15), n = lane & 15, kh = (lane >> 4) << 1;
  v8f acc = {};
  for (int k = 0; k < 16; k += 4) {
    int kb = k + kh;
    v2f a, b;
    a.x = A[(size_t)m * 16 + kb];   a.y = A[(size_t)m * 16 + kb + 1];
    b.x = Wa[kb * 16 + n];          b.y = Wa[(kb + 1) * 16 + n];
    acc = wmma4(a, b, acc);
    if (A2) {
      a.x = A2[(size_t)m * 16 + kb]; a.y = A2[(size_t)m * 16 + kb + 1];
      b.x = Wb[kb * 16 + n];         b.y = Wb[(kb + 1) * 16 + n];
      acc = wmma4(a, b, acc);
    }
  }
  float bn = bias ? bias[n] : 0.f;
  int mb = tm + ((lane >> 4) << 3);
#pragma unroll
  for (int i = 0; i < 8; ++i) {
    float v = acc[i] + bn;
    if (do_relu) v = v > 0.f ? v : 0.f;
    C[(size_t)(mb + i) * 16 + (lane & 15)] = v;
  }
}

// ---------------- pairwise MLP update: S_hat += relu(Ps - Pt) . Wm2 + bm2 ----
__global__ void k_update(const float* __restrict__ Ps, const float* __restrict__ Pt,
                         const float* __restrict__ Wm2, const float* __restrict__ bm2,
                         float* __restrict__ Shat) {
  __shared__ float ps[16], w2[16];
  int blk = blockIdx.x;                 // 16384 blocks: (b*N + s) x 4 t-segments
  int tseg = blk & 3;
  int srow = blk >> 2;                  // b*1024 + s
  if (threadIdx.x < 16) {
    ps[threadIdx.x] = Ps[(size_t)srow * 16 + threadIdx.x];
    w2[threadIdx.x] = Wm2[threadIdx.x];
  }
  __syncthreads();
  int b = srow >> 10;
  int t = tseg * 256 + threadIdx.x;
  const float* pt = Pt + ((size_t)(b * Nn + t)) * 16;
  float acc = bm2[0];
#pragma unroll
  for (int r = 0; r < 16; ++r) {
    float d = ps[r] - pt[r];
    acc += (d > 0.f ? d : 0.f) * w2[r];
  }
  Shat[(size_t)srow * Nn + t] += acc;
}

// ---------------- host orchestration ----------------
extern "C" void kernel_launch(void* const* d_in, const int* in_sizes, int n_in,
                              void* d_out, int out_size, void* d_ws, size_t ws_size,
                              hipStream_t stream) {
  const float* x_s     = (const float*)d_in[0];
  const int*   ei_s    = (const int*)  d_in[1];
  const float* x_t     = (const float*)d_in[2];
  const int*   ei_t    = (const int*)  d_in[3];
  const float* W1s     = (const float*)d_in[4];
  const float* W1n     = (const float*)d_in[5];
  const float* b1      = (const float*)d_in[6];
  const float* W2s     = (const float*)d_in[7];
  const float* W2n     = (const float*)d_in[8];
  const float* b2      = (const float*)d_in[9];
  const float* Wm1     = (const float*)d_in[10];
  const float* bm1     = (const float*)d_in[11];
  const float* Wm2     = (const float*)d_in[12];
  const float* bm2     = (const float*)d_in[13];
  const float* r_steps = (const float*)d_in[14];
  float* out = (float*)d_out;

  // workspace layout (floats)
  float* w = (float*)d_ws;
  float* agg_s = w;                    // 524288
  float* agg_t = agg_s + 524288;       // 524288
  float* h_s   = agg_t + 524288;       // 524288
  float* h_t   = h_s + 524288;         // 524288
  float* Shat  = h_t + 524288;         // 4194304
  float* mx    = Shat + SHATSZ;        // 4096
  float* sinv  = mx + 4096;            // 4096
  float* rt    = sinv + 4096;          // 65536
  float* agr_s = rt + 65536;           // 65536
  float* agr_t = agr_s + 65536;        // 65536
  float* o_s   = agr_t + 65536;        // 65536
  float* o_t   = o_s + 65536;          // 65536
  float* P_s   = o_t + 65536;          // 65536
  float* P_t   = P_s + 65536;          // 65536

  dim3 blk(256);

  // --- psi_1 on both graphs ---
  k_zero<<<2048, blk, 0, stream>>>(agg_s, 524288);
  k_zero<<<2048, blk, 0, stream>>>(agg_t, 524288);
  k_scatter128<<<32768, blk, 0, stream>>>(x_s, ei_s, agg_s);
  k_scatter128<<<32768, blk, 0, stream>>>(x_t, ei_t, agg_t);
  k_psi1_gemm<<<256, blk, 0, stream>>>(x_s, agg_s, W1s, W1n, b1, h_s);
  k_psi1_gemm<<<256, blk, 0, stream>>>(x_t, agg_t, W1s, W1n, b1, h_t);

  // --- S_hat = h_s @ h_t^T (LDS-tiled, async staged) ---
  k_shat_gemm<<<1024, blk, 0, stream>>>(h_s, h_t, Shat);

  // --- S_0 output ---
  k_row_stats<<<4096, blk, 0, stream>>>(Shat, mx, sinv);
  k_softmax_out<<<16384, blk, 0, stream>>>(Shat, mx, sinv, out);

  // --- refinement steps ---
  for (int i = 0; i < NSTEPS; ++i) {
    const float* rs_i = r_steps + (size_t)i * BN * Rr;
    if (i > 0) k_row_stats<<<4096, blk, 0, stream>>>(Shat, mx, sinv);
    // r_t = softmax(S_hat)^T @ r_s
    k_rt_gemm<<<32, blk, 0, stream>>>(Shat, mx, sinv, rs_i, rt);
    // psi_2 on graph s (input r_s) and graph t (input r_t)
    k_zero<<<256, blk, 0, stream>>>(agr_s, 65536);
    k_zero<<<256, blk, 0, stream>>>(agr_t, 65536);
    k_scatter16<<<4096, blk, 0, stream>>>(rs_i, ei_s, agr_s);
    k_scatter16<<<4096, blk, 0, stream>>>(rt,   ei_t, agr_t);
    k_gemm16<<<32, blk, 0, stream>>>(rs_i, W2s, agr_s, W2n, b2, o_s, 1);
    k_gemm16<<<32, blk, 0, stream>>>(rt,   W2s, agr_t, W2n, b2, o_t, 1);
    // P_s = o_s@Wm1 + bm1 ;  P_t = o_t@Wm1   (relu(D@Wm1+bm1) = relu(P_s - P_t))
    k_gemm16<<<32, blk, 0, stream>>>(o_s, Wm1, nullptr, nullptr, bm1, P_s, 0);
    k_gemm16<<<32, blk, 0, stream>>>(o_t, Wm1, nullptr, nullptr, nullptr, P_t, 0);
    // S_hat += relu(P_s - P_t) . Wm2 + bm2
    k_update<<<16384, blk, 0, stream>>>(P_s, P_t, Wm2, bm2, Shat);
  }

  // --- S_L output ---
  k_row_stats<<<4096, blk, 0, stream>>>(Shat, mx, sinv);
  k_softmax_out<<<16384, blk, 0, stream>>>(Shat, mx, sinv, out + SHATSZ);
}